// SpatialEncoder_44169443672094
// MI455X (gfx1250) — compile-verified
//
#include <hip/hip_runtime.h>

#define Nn 100000
#define Ee 1600000
#define Bb 256
#define DOUT 48
#define NEG_SLOPE 0.2f
#define LN_EPS 1e-5f

typedef __attribute__((ext_vector_type(2))) float v2f;
typedef __attribute__((ext_vector_type(8))) float v8f;

// ---------------------------------------------------------------- helpers
__device__ __forceinline__ float lrelu(float x) { return x > 0.f ? x : NEG_SLOPE * x; }
__device__ __forceinline__ float eluf(float x)  { return x > 0.f ? x : expm1f(x); }

// IEEE-ordered float atomic max via int/uint atomics (resolved in L2).
__device__ __forceinline__ void atomicMaxF(float* addr, float v) {
  if (v >= 0.f) atomicMax((int*)addr, __float_as_int(v));
  else          atomicMin((unsigned int*)addr, __float_as_uint(v));
}

#define WMMA_F32X4(A, B, C) \
  __builtin_amdgcn_wmma_f32_16x16x4_f32(false, (A), false, (B), (short)0, (C), false, false)

// ---------------------------------------------------------------- fill
__global__ void k_fill(float* __restrict__ p, float v, int n) {
  int i = blockIdx.x * blockDim.x + threadIdx.x;
  if (i < n) p[i] = v;
}

// ---------------------------------------------------------------- GEMM1: h = x @ W1  [N,3]@[3,128] (K padded to 4)
__global__ void __launch_bounds__(256) k_gemm1(const float* __restrict__ x,
                                               const float* __restrict__ W1,
                                               float* __restrict__ H) {
  int lane = threadIdx.x & 31;
  int nb = (blockIdx.x * (blockDim.x >> 5) + (threadIdx.x >> 5)) * 16;
  if (nb >= Nn) return;                       // wave-uniform
  int m = lane & 15;
  bool lo = lane < 16;
  int node = min(nb + m, Nn - 1);
  v2f a;
  if (lo) { a.x = x[node * 3 + 0]; a.y = x[node * 3 + 1]; }
  else    { a.x = x[node * 3 + 2]; a.y = 0.f; }
  int rbase = nb + (lo ? 0 : 8);
  bool full = (nb + 16 <= Nn);                // wave-uniform fast path
#pragma unroll
  for (int nt = 0; nt < 8; ++nt) {
    int col = nt * 16 + m;
    v2f b;
    if (lo) { b.x = W1[col];       b.y = W1[128 + col]; }
    else    { b.x = W1[256 + col]; b.y = 0.f; }
    v8f acc = {};
    acc = WMMA_F32X4(a, b, acc);
    if (full) {
#pragma unroll
      for (int r = 0; r < 8; ++r) H[(size_t)(rbase + r) * 128 + col] = acc[r];
    } else {
#pragma unroll
      for (int r = 0; r < 8; ++r) {
        int row = rbase + r;
        if (row < Nn) H[(size_t)row * 128 + col] = acc[r];
      }
    }
  }
}

// ---------------------------------------------------------------- GEMM2: h2pre = h1 @ W2  [N,128]@[128,64], W2 in LDS
__global__ void __launch_bounds__(256) k_gemm2(const float* __restrict__ h1,
                                               const float* __restrict__ W2,
                                               float* __restrict__ out) {
  __shared__ float sW[128 * 64];
  for (int i = threadIdx.x; i < 128 * 64; i += 256) sW[i] = W2[i];
  __syncthreads();
  int lane = threadIdx.x & 31;
  int nb = (blockIdx.x * 8 + (threadIdx.x >> 5)) * 16;
  int m = lane & 15;
  bool lo = lane < 16;
  int node = min(nb + m, Nn - 1);
  const float* arow = h1 + (size_t)node * 128 + (lo ? 0 : 2);
  v8f acc0 = {}, acc1 = {}, acc2 = {}, acc3 = {};
  for (int k = 0; k < 32; ++k) {
    v2f a; a.x = arow[4 * k]; a.y = arow[4 * k + 1];
    const float* bp0 = &sW[(4 * k + (lo ? 0 : 2)) * 64 + m];
    v2f b;
    b.x = bp0[0];  b.y = bp0[64];       acc0 = WMMA_F32X4(a, b, acc0);
    b.x = bp0[16]; b.y = bp0[64 + 16];  acc1 = WMMA_F32X4(a, b, acc1);
    b.x = bp0[32]; b.y = bp0[64 + 32];  acc2 = WMMA_F32X4(a, b, acc2);
    b.x = bp0[48]; b.y = bp0[64 + 48];  acc3 = WMMA_F32X4(a, b, acc3);
  }
  int rbase = nb + (lo ? 0 : 8);
  bool full = (nb + 16 <= Nn);                // wave-uniform fast path
  if (full) {
#pragma unroll
    for (int r = 0; r < 8; ++r) {
      float* o = out + (size_t)(rbase + r) * 64 + m;
      o[0] = acc0[r]; o[16] = acc1[r]; o[32] = acc2[r]; o[48] = acc3[r];
    }
  } else {
#pragma unroll
    for (int r = 0; r < 8; ++r) {
      int row = rbase + r;
      if (row < Nn) {
        float* o = out + (size_t)row * 64 + m;
        o[0] = acc0[r]; o[16] = acc1[r]; o[32] = acc2[r]; o[48] = acc3[r];
      }
    }
  }
}

// ---------------------------------------------------------------- attention coefficients
template <int HEADS>
__global__ void k_att(const float* __restrict__ feat, const float* __restrict__ asrc,
                      const float* __restrict__ adst, float* __restrict__ es,
                      float* __restrict__ ed) {
  int i = blockIdx.x * blockDim.x + threadIdx.x;
  if (i >= Nn * HEADS) return;
  int n = i / HEADS, h = i % HEADS;
  const float* f = feat + (size_t)n * (HEADS * 32) + h * 32;
  const float* a = asrc + h * 32;
  const float* b = adst + h * 32;
  float s0 = 0.f, s1 = 0.f;
#pragma unroll
  for (int c = 0; c < 32; ++c) { s0 += f[c] * a[c]; s1 += f[c] * b[c]; }
  es[i] = s0; ed[i] = s1;
}

// ---------------------------------------------------------------- edge pass: segment max / exp-sum
template <int HEADS, bool DO_MAX>
__global__ void k_edge_pass(const int* __restrict__ src, const int* __restrict__ dst,
                            const float* __restrict__ es, const float* __restrict__ ed,
                            float* __restrict__ m, float* __restrict__ s) {
  int i = blockIdx.x * blockDim.x + threadIdx.x;
  if (i >= Ee + Nn) return;
  int si, di;
  if (i < Ee) { si = src[i]; di = dst[i]; } else { si = i - Ee; di = si; }
#pragma unroll
  for (int h = 0; h < HEADS; ++h) {
    float e = lrelu(es[si * HEADS + h] + ed[di * HEADS + h]);
    if (DO_MAX) atomicMaxF(&m[di * HEADS + h], e);
    else        atomicAdd(&s[di * HEADS + h], expf(e - m[di * HEADS + h]));
  }
}

// ---------------------------------------------------------------- edge pass: weighted message scatter (wave per edge)
template <int CH, int HEADS>
__global__ void k_edge_msg(const int* __restrict__ src, const int* __restrict__ dst,
                           const float* __restrict__ feat, const float* __restrict__ es,
                           const float* __restrict__ ed, const float* __restrict__ m,
                           const float* __restrict__ s, float* __restrict__ out) {
  int wave = blockIdx.x * (blockDim.x >> 5) + (threadIdx.x >> 5);
  if (wave >= Ee + Nn) return;
  int lane = threadIdx.x & 31;
  int si, di;
  if (wave < Ee) { si = src[wave]; di = dst[wave]; } else { si = wave - Ee; di = si; }
  constexpr int PER = CH / 32;
  int c0 = lane * PER;
  int head = c0 / (CH / HEADS);
  float e = lrelu(es[si * HEADS + head] + ed[di * HEADS + head]);
  float alpha = expf(e - m[di * HEADS + head]) / (s[di * HEADS + head] + 1e-16f);
  const float* f = feat + (size_t)si * CH + c0;
  float* o = out + (size_t)di * CH + c0;
  if constexpr (PER == 4) {
    float4 fv = *(const float4*)f;               // global_load_b128
    atomicAdd(&o[0], fv.x * alpha);
    atomicAdd(&o[1], fv.y * alpha);
    atomicAdd(&o[2], fv.z * alpha);
    atomicAdd(&o[3], fv.w * alpha);
  } else {
    float2 fv = *(const float2*)f;               // global_load_b64
    atomicAdd(&o[0], fv.x * alpha);
    atomicAdd(&o[1], fv.y * alpha);
  }
}

// ---------------------------------------------------------------- layer1 post: bias + ELU + LayerNorm(128), in place
__global__ void __launch_bounds__(256) k_post1(float* __restrict__ h, const float* __restrict__ b1,
                                               const float* __restrict__ g1,
                                               const float* __restrict__ be1) {
  int n = blockIdx.x * (blockDim.x >> 5) + (threadIdx.x >> 5);
  if (n >= Nn) return;
  int lane = threadIdx.x & 31;
  float v[4]; float sum = 0.f;
#pragma unroll
  for (int j = 0; j < 4; ++j) {
    int c = lane * 4 + j;
    float t = eluf(h[(size_t)n * 128 + c] + b1[c]);
    v[j] = t; sum += t;
  }
#pragma unroll
  for (int o = 16; o > 0; o >>= 1) sum += __shfl_xor(sum, o, 32);
  float mean = sum * (1.f / 128.f);
  float var = 0.f;
#pragma unroll
  for (int j = 0; j < 4; ++j) { float d = v[j] - mean; var += d * d; }
#pragma unroll
  for (int o = 16; o > 0; o >>= 1) var += __shfl_xor(var, o, 32);
  float rs = rsqrtf(var * (1.f / 128.f) + LN_EPS);
#pragma unroll
  for (int j = 0; j < 4; ++j) {
    int c = lane * 4 + j;
    h[(size_t)n * 128 + c] = (v[j] - mean) * rs * g1[c] + be1[c];
  }
}

// ---------------------------------------------------------------- layer2 post: head mean + b2 + ELU + skip + LN(32)
__global__ void __launch_bounds__(256) k_post2(const float* __restrict__ out2,
                                               const float* __restrict__ x,
                                               const float* __restrict__ b2,
                                               const float* __restrict__ Wsk,
                                               const float* __restrict__ bsk,
                                               const float* __restrict__ g2,
                                               const float* __restrict__ be2,
                                               float* __restrict__ h2) {
  int n = blockIdx.x * (blockDim.x >> 5) + (threadIdx.x >> 5);
  if (n >= Nn) return;
  int c = threadIdx.x & 31;
  float v = 0.5f * (out2[(size_t)n * 64 + c] + out2[(size_t)n * 64 + 32 + c]) + b2[c];
  v = eluf(v);
  v += x[n * 3 + 0] * Wsk[c] + x[n * 3 + 1] * Wsk[32 + c] + x[n * 3 + 2] * Wsk[64 + c] + bsk[c];
  float sum = v;
#pragma unroll
  for (int o = 16; o > 0; o >>= 1) sum += __shfl_xor(sum, o, 32);
  float mean = sum * (1.f / 32.f);
  float d = v - mean, var = d * d;
#pragma unroll
  for (int o = 16; o > 0; o >>= 1) var += __shfl_xor(var, o, 32);
  float rs = rsqrtf(var * (1.f / 32.f) + LN_EPS);
  h2[(size_t)n * 32 + c] = (v - mean) * rs * g2[c] + be2[c];
}

// ---------------------------------------------------------------- pooling pass A: sum, max, count
__global__ void k_pool_a(const float* __restrict__ h2, const int* __restrict__ batch,
                         float* __restrict__ msum, float* __restrict__ mmax,
                         float* __restrict__ cnt) {
  int i = blockIdx.x * blockDim.x + threadIdx.x;
  if (i >= Nn * 32) return;
  int n = i >> 5, c = i & 31;
  int b = batch[n];
  float v = h2[i];
  atomicAdd(&msum[b * 32 + c], v);
  atomicMaxF(&mmax[b * 32 + c], v);
  if (c == 0) atomicAdd(&cnt[b], 1.f);
}

// ---------------------------------------------------------------- pooling pass B: centered sum of squares
__global__ void k_pool_b(const float* __restrict__ h2, const int* __restrict__ batch,
                         const float* __restrict__ msum, const float* __restrict__ cnt,
                         float* __restrict__ sq) {
  int i = blockIdx.x * blockDim.x + threadIdx.x;
  if (i >= Nn * 32) return;
  int n = i >> 5, c = i & 31;
  int b = batch[n];
  float cn = fmaxf(cnt[b], 1.f);
  float d = h2[i] - msum[b * 32 + c] / cn;
  atomicAdd(&sq[b * 32 + c], d * d);
}

// ---------------------------------------------------------------- final: [mean|max|std] @ Wp + bp
__global__ void k_final(const float* __restrict__ msum, const float* __restrict__ mmax,
                        const float* __restrict__ sq, const float* __restrict__ cnt,
                        const float* __restrict__ Wp, const float* __restrict__ bp,
                        float* __restrict__ out) {
  int i = blockIdx.x * blockDim.x + threadIdx.x;
  if (i >= Bb * DOUT) return;
  int g = i / DOUT, j = i % DOUT;
  float cn = fmaxf(cnt[g], 1.f);
  float acc = bp[j];
#pragma unroll
  for (int c = 0; c < 32; ++c) {
    acc += (msum[g * 32 + c] / cn)        * Wp[c * DOUT + j];
    acc += mmax[g * 32 + c]               * Wp[(32 + c) * DOUT + j];
    acc += sqrtf(sq[g * 32 + c] / cn)     * Wp[(64 + c) * DOUT + j];
  }
  out[i] = acc;
}

// ---------------------------------------------------------------- launch
static inline int cdiv(long a, long b) { return (int)((a + b - 1) / b); }

extern "C" void kernel_launch(void* const* d_in, const int* in_sizes, int n_in,
                              void* d_out, int out_size, void* d_ws, size_t ws_size,
                              hipStream_t stream) {
  const float* x   = (const float*)d_in[0];
  const int*   ei  = (const int*)d_in[1];
  const int*   bat = (const int*)d_in[2];
  const float* W1  = (const float*)d_in[3];
  const float* as1 = (const float*)d_in[4];
  const float* ad1 = (const float*)d_in[5];
  const float* b1  = (const float*)d_in[6];
  const float* W2  = (const float*)d_in[7];
  const float* as2 = (const float*)d_in[8];
  const float* ad2 = (const float*)d_in[9];
  const float* b2  = (const float*)d_in[10];
  const float* Wsk = (const float*)d_in[11];
  const float* bsk = (const float*)d_in[12];
  const float* g1  = (const float*)d_in[13];
  const float* be1 = (const float*)d_in[14];
  const float* g2  = (const float*)d_in[15];
  const float* be2 = (const float*)d_in[16];
  const float* Wp  = (const float*)d_in[17];
  const float* bp  = (const float*)d_in[18];
  float* out = (float*)d_out;
  const int* srcv = ei;
  const int* dstv = ei + Ee;

  // workspace layout (floats); total ~312*N + 25K  ~= 125 MB
  float* F    = (float*)d_ws;
  float* H1F  = F;                          // N*128  (layer1 features h)
  float* OUT1 = H1F + (size_t)Nn * 128;     // N*128  (layer1 accum -> h1)
  float* ES1  = OUT1 + (size_t)Nn * 128;    // N*4
  float* ED1  = ES1 + Nn * 4;               // N*4
  float* M1   = ED1 + Nn * 4;               // N*4
  float* S1   = M1 + Nn * 4;                // N*4
  float* ES2  = S1 + Nn * 4;                // N*2
  float* ED2  = ES2 + Nn * 2;               // N*2
  float* M2   = ED2 + Nn * 2;               // N*2
  float* S2   = M2 + Nn * 2;                // N*2
  float* H2   = S2 + Nn * 2;                // N*32
  float* MSUM = H2 + (size_t)Nn * 32;       // B*32
  float* MMAX = MSUM + Bb * 32;             // B*32
  float* SQ   = MMAX + Bb * 32;             // B*32
  float* CNT  = SQ + Bb * 32;               // B
  // aliases (H1F dead after layer-1 message pass)
  float* H2PRE = H1F;                       // N*64
  float* OUT2  = H1F + (size_t)Nn * 64;     // N*64

  const float NINF = -__builtin_huge_valf();
  const int T = 256;

  // ---- init
  k_fill<<<cdiv((long)Nn * 128, T), T, 0, stream>>>(OUT1, 0.f, Nn * 128);
  k_fill<<<cdiv(Nn * 4, T), T, 0, stream>>>(M1, NINF, Nn * 4);
  k_fill<<<cdiv(Nn * 4, T), T, 0, stream>>>(S1, 0.f, Nn * 4);
  k_fill<<<cdiv(Nn * 2, T), T, 0, stream>>>(M2, NINF, Nn * 2);
  k_fill<<<cdiv(Nn * 2, T), T, 0, stream>>>(S2, 0.f, Nn * 2);
  k_fill<<<cdiv(Bb * 32, T), T, 0, stream>>>(MSUM, 0.f, Bb * 32);
  k_fill<<<cdiv(Bb * 32, T), T, 0, stream>>>(MMAX, NINF, Bb * 32);
  k_fill<<<cdiv(Bb * 32, T), T, 0, stream>>>(SQ, 0.f, Bb * 32);
  k_fill<<<cdiv(Bb, T), T, 0, stream>>>(CNT, 0.f, Bb);

  const int EN = Ee + Nn;

  // ---- layer 1
  k_gemm1<<<cdiv(Nn, 128), T, 0, stream>>>(x, W1, H1F);
  k_att<4><<<cdiv((long)Nn * 4, T), T, 0, stream>>>(H1F, as1, ad1, ES1, ED1);
  k_edge_pass<4, true><<<cdiv(EN, T), T, 0, stream>>>(srcv, dstv, ES1, ED1, M1, S1);
  k_edge_pass<4, false><<<cdiv(EN, T), T, 0, stream>>>(srcv, dstv, ES1, ED1, M1, S1);
  k_edge_msg<128, 4><<<cdiv(EN, 8), T, 0, stream>>>(srcv, dstv, H1F, ES1, ED1, M1, S1, OUT1);
  k_post1<<<cdiv(Nn, 8), T, 0, stream>>>(OUT1, b1, g1, be1);

  // H1F is now dead -> reuse for layer 2; zero OUT2 alias region
  k_fill<<<cdiv((long)Nn * 64, T), T, 0, stream>>>(OUT2, 0.f, Nn * 64);

  // ---- layer 2
  k_gemm2<<<cdiv(Nn, 128), T, 0, stream>>>(OUT1, W2, H2PRE);
  k_att<2><<<cdiv((long)Nn * 2, T), T, 0, stream>>>(H2PRE, as2, ad2, ES2, ED2);
  k_edge_pass<2, true><<<cdiv(EN, T), T, 0, stream>>>(srcv, dstv, ES2, ED2, M2, S2);
  k_edge_pass<2, false><<<cdiv(EN, T), T, 0, stream>>>(srcv, dstv, ES2, ED2, M2, S2);
  k_edge_msg<64, 2><<<cdiv(EN, 8), T, 0, stream>>>(srcv, dstv, H2PRE, ES2, ED2, M2, S2, OUT2);
  k_post2<<<cdiv(Nn, 8), T, 0, stream>>>(OUT2, x, b2, Wsk, bsk, g2, be2, H2);

  // ---- pooling + projection
  k_pool_a<<<cdiv((long)Nn * 32, T), T, 0, stream>>>(H2, bat, MSUM, MMAX, CNT);
  k_pool_b<<<cdiv((long)Nn * 32, T), T, 0, stream>>>(H2, bat, MSUM, CNT, SQ);
  k_final<<<cdiv(Bb * DOUT, T), T, 0, stream>>>(MSUM, MMAX, SQ, CNT, Wp, bp, out);
}